// RNN_39024072851620
// MI455X (gfx1250) — compile-verified
//
#include <hip/hip_runtime.h>
#include <math.h>

// ---------------------------------------------------------------------------
// 2-layer RNN (E=H=O=1024, B=64, T=512) for MI455X (gfx1250, wave32).
// fp32 WMMA (V_WMMA_F32_16X16X4_F32) for bit-fidelity with the fp32 reference.
//
// Phase 1: input-projection GEMM (32768x1024x1024), 1 wave / 16x16 tile,
//          4 independent accumulator chains (plenty of TLP there).
// Phase 2: persistent kernel, 64 blocks (one per WGP). Block owns one
//          16-column slab of all four recurrent weight matrices, preloaded
//          ONCE into LDS transposed (257 KB <= 320 KB/WGP), so the 512-step
//          loop reads B-fragments as conflict-free ds_load_b64. 8 independent
//          accumulator chains per GEMM (recurrence is chain-latency bound at
//          1 wave/SIMD). x_proj tile for step t+1 is prefetched during step t
//          (global_prefetch_b8). 2 device-wide barriers per step.
// ---------------------------------------------------------------------------

typedef __attribute__((ext_vector_type(2))) float v2f;
typedef __attribute__((ext_vector_type(8))) float v8f;

#define RNN_B 64
#define RNN_T 512
#define RNN_H 1024     // E == H == O
#define REC_BLOCKS 64
#define SLAB_LD 1028   // 1024 + 4 pad floats -> lane stride 16B mod 256B:
                       // bank = (4n + 2*half + k) % 64, all 64 banks hit once
#define SLAB_FLOATS (16 * SLAB_LD)
#define REC_LDS_BYTES (4u * SLAB_FLOATS * sizeof(float))  // 263168 B

__device__ __forceinline__ v8f wmma4(v2f a, v2f b, v8f c) {
  return __builtin_amdgcn_wmma_f32_16x16x4_f32(false, a, false, b,
                                               (short)0, c, false, false);
}

// ---- GEMM tile, A + B both global, 4 accumulator chains (phase-1 only) ----
// A: [16 x 1024] row-major (lda=1024); Bm: [1024 x 16] col slab (ldb=1024)
__device__ __forceinline__ v8f tile_gemm_global4(const float* __restrict__ A,
                                                 const float* __restrict__ Bm) {
  const int lane = threadIdx.x & 31;
  const int half = lane >> 4;
  const int mn   = lane & 15;
  const float* ap = A + (size_t)mn * RNN_H + 2 * half;
  const float* bp = Bm + mn + (size_t)(2 * half) * RNN_H;
  v8f a0 = {}, a1 = {}, a2 = {}, a3 = {};
#pragma unroll 4
  for (int k = 0; k < 256; k += 4) {
    v2f b;
    b.x = bp[(size_t)k * RNN_H];
    b.y = bp[(size_t)k * RNN_H + RNN_H];
    a0 = wmma4(*(const v2f*)(ap + k), b, a0);
    b.x = bp[(size_t)(k + 256) * RNN_H];
    b.y = bp[(size_t)(k + 256) * RNN_H + RNN_H];
    a1 = wmma4(*(const v2f*)(ap + k + 256), b, a1);
    b.x = bp[(size_t)(k + 512) * RNN_H];
    b.y = bp[(size_t)(k + 512) * RNN_H + RNN_H];
    a2 = wmma4(*(const v2f*)(ap + k + 512), b, a2);
    b.x = bp[(size_t)(k + 768) * RNN_H];
    b.y = bp[(size_t)(k + 768) * RNN_H + RNN_H];
    a3 = wmma4(*(const v2f*)(ap + k + 768), b, a3);
  }
  return (a0 + a1) + (a2 + a3);
}

// ---- GEMM tile, A global, B transposed in LDS, 8 accumulator chains -------
// slabT[n*SLAB_LD + k] == W[k][col0 + n]  ->  B-frag is one ds_load_b64
__device__ __forceinline__ v8f tile_gemm_lds8(const float* __restrict__ A,
                                              const float* __restrict__ slabT) {
  const int lane = threadIdx.x & 31;
  const int half = lane >> 4;
  const int mn   = lane & 15;
  const float* ap = A + (size_t)mn * RNN_H + 2 * half;
  const float* bp = slabT + mn * SLAB_LD + 2 * half;
  v8f acc[8] = {};
#pragma unroll 2
  for (int k = 0; k < 128; k += 4) {
#pragma unroll
    for (int c = 0; c < 8; ++c) {
      acc[c] = wmma4(*(const v2f*)(ap + k + c * 128),
                     *(const v2f*)(bp + k + c * 128), acc[c]);
    }
  }
  return ((acc[0] + acc[1]) + (acc[2] + acc[3])) +
         ((acc[4] + acc[5]) + (acc[6] + acc[7]));
}

// ---- fused dual GEMM: A0@S0 + A1@S1, 4 chains per matrix ------------------
__device__ __forceinline__ v8f tile_gemm2_lds8(const float* __restrict__ A0,
                                               const float* __restrict__ S0,
                                               const float* __restrict__ A1,
                                               const float* __restrict__ S1) {
  const int lane = threadIdx.x & 31;
  const int half = lane >> 4;
  const int mn   = lane & 15;
  const float* a0p = A0 + (size_t)mn * RNN_H + 2 * half;
  const float* a1p = A1 + (size_t)mn * RNN_H + 2 * half;
  const float* b0p = S0 + mn * SLAB_LD + 2 * half;
  const float* b1p = S1 + mn * SLAB_LD + 2 * half;
  v8f acc[8] = {};
#pragma unroll 2
  for (int k = 0; k < 256; k += 4) {
#pragma unroll
    for (int c = 0; c < 4; ++c) {
      acc[c]     = wmma4(*(const v2f*)(a0p + k + c * 256),
                         *(const v2f*)(b0p + k + c * 256), acc[c]);
      acc[4 + c] = wmma4(*(const v2f*)(a1p + k + c * 256),
                         *(const v2f*)(b1p + k + c * 256), acc[4 + c]);
    }
  }
  return ((acc[0] + acc[1]) + (acc[2] + acc[3])) +
         ((acc[4] + acc[5]) + (acc[6] + acc[7]));
}

// ---- cooperative transposed weight-slab preload (once per kernel) ---------
// Wcol points at W[0][col0]; writes slabT[n][k] = W[k][col0+n], 128 threads.
__device__ __forceinline__ void preload_slabT(const float* __restrict__ Wcol,
                                              float* __restrict__ slabT) {
  const int n  = threadIdx.x & 15;
  const int k0 = threadIdx.x >> 4;    // 0..7
#pragma unroll 8
  for (int k = k0; k < 1024; k += 8)
    slabT[n * SLAB_LD + k] = Wcol[(size_t)k * RNN_H + n];
}

// ---- device-wide sense barrier (agent scope) ------------------------------
// All threads run __threadfence() on exit: agent-scope acquire per wave
// (global_inv) so no WGP cache serves stale ping-pong state.
__device__ __forceinline__ void grid_barrier(unsigned* bar, unsigned nblk) {
  __threadfence();
  __syncthreads();
  if (threadIdx.x == 0) {
    unsigned* arrive = bar;
    unsigned* gen    = bar + 64;   // separate cache line
    unsigned g = __hip_atomic_load(gen, __ATOMIC_RELAXED,
                                   __HIP_MEMORY_SCOPE_AGENT);
    unsigned a = __hip_atomic_fetch_add(arrive, 1u, __ATOMIC_ACQ_REL,
                                        __HIP_MEMORY_SCOPE_AGENT);
    if (a == nblk - 1u) {
      __hip_atomic_store(arrive, 0u, __ATOMIC_RELAXED,
                         __HIP_MEMORY_SCOPE_AGENT);
      __hip_atomic_fetch_add(gen, 1u, __ATOMIC_RELEASE,
                             __HIP_MEMORY_SCOPE_AGENT);
    } else {
      while (__hip_atomic_load(gen, __ATOMIC_ACQUIRE,
                               __HIP_MEMORY_SCOPE_AGENT) == g) {
        __builtin_amdgcn_s_sleep(2);
      }
    }
  }
  __syncthreads();
  __threadfence();
}

// ---- zero hidden-state ping-pong buffers + barrier state ------------------
__global__ void rnn_init_kernel(float* __restrict__ hbufs,
                                unsigned* __restrict__ bar) {
  size_t i = (size_t)blockIdx.x * blockDim.x + threadIdx.x;
  const size_t n = 4ull * RNN_B * RNN_H;  // h0 x2 + h1 x2
  if (i < n) hbufs[i] = 0.0f;
  if (i < 128) bar[i] = 0u;
}

// ---- Phase 1: x_proj[t,b,h] = emb[b,t,:] @ W_ih + b_ih --------------------
__global__ void __launch_bounds__(256)
rnn_xproj_kernel(const float* __restrict__ emb, const float* __restrict__ Wih,
                 const float* __restrict__ bih, float* __restrict__ xp) {
  const int wglob = blockIdx.x * 8 + (threadIdx.x >> 5);
  const int tm = wglob >> 6;   // row tile over r = b*T + t  (0..2047)
  const int tn = wglob & 63;   // col tile over H            (0..63)
  const int lane = threadIdx.x & 31;
  const int half = lane >> 4;
  const int mn   = lane & 15;

  v8f acc = tile_gemm_global4(emb + (size_t)tm * 16 * RNN_H, Wih + tn * 16);

  const int colg = tn * 16 + mn;
  const float bias = bih[colg];
#pragma unroll
  for (int v = 0; v < 8; ++v) {
    int r = tm * 16 + v + 8 * half;     // r = b*T + t (emb is [B,T,E])
    int b = r >> 9;                     // T = 512
    int t = r & (RNN_T - 1);
    xp[((size_t)t * RNN_B + b) * RNN_H + colg] = acc[v] + bias;
  }
}

// ---- Phase 2: persistent recurrent kernel ---------------------------------
// blockIdx.x = hj (column slab 0..63); the 4 waves = bi (batch-row 0..3).
// All weight slabs for this block live in LDS for the entire 512-step loop.
__global__ void __launch_bounds__(128)
rnn_recurrent_kernel(const float* __restrict__ xp,
                     const float* __restrict__ Whh, const float* __restrict__ bhh,
                     const float* __restrict__ Wl,  const float* __restrict__ bl,
                     const float* __restrict__ Wr,  const float* __restrict__ br,
                     const float* __restrict__ Who, const float* __restrict__ bho,
                     float* __restrict__ h0buf, float* __restrict__ h1buf,
                     unsigned* __restrict__ bar, float* __restrict__ out) {
  extern __shared__ float smem[];
  float* slab_hh = smem;                    // W_hh[:, hj*16 .. +16] transposed
  float* slab_l  = smem + 1 * SLAB_FLOATS;  // W_l
  float* slab_r  = smem + 2 * SLAB_FLOATS;  // W_r
  float* slab_ho = smem + 3 * SLAB_FLOATS;  // W_ho

  const int hj = blockIdx.x;          // column block (0..63)
  const int bi = threadIdx.x >> 5;    // batch-row block (0..3)
  const int lane = threadIdx.x & 31;
  const int half = lane >> 4;
  const int mn   = lane & 15;
  const int colg = hj * 16 + mn;
  const int rowg = bi * 16;

  // one-time transposed preload of the four weight slabs (256 KB / block)
  preload_slabT(Whh + hj * 16, slab_hh);
  preload_slabT(Wl  + hj * 16, slab_l);
  preload_slabT(Wr  + hj * 16, slab_r);
  preload_slabT(Who + hj * 16, slab_ho);
  __syncthreads();

  const float bias_hh = bhh[colg];
  const float bias_lr = bl[colg] + br[colg];
  const float bias_ho = bho[colg];
  const size_t BH = (size_t)RNN_B * RNN_H;

  // warm the first x_proj tile (each lane: one 64B line of its row)
  __builtin_prefetch(xp + ((size_t)rowg + mn) * RNN_H + hj * 16, 0, 1);

  for (int t = 0; t < RNN_T; ++t) {
    const float* h0_old = h0buf + (size_t)(t & 1) * BH;
    float*       h0_new = h0buf + (size_t)((t + 1) & 1) * BH;
    const float* h1_old = h1buf + (size_t)(t & 1) * BH;
    float*       h1_new = h1buf + (size_t)((t + 1) & 1) * BH;

    // prefetch next step's x_proj tile (a full step of lead time; xp streams
    // from HBM since xp + out together exceed L2)
    if (t + 1 < RNN_T) {
      __builtin_prefetch(
          xp + (((size_t)(t + 1) * RNN_B) + rowg + mn) * RNN_H + hj * 16,
          0, 1);
    }

    // --- layer 0: h0 = tanh(xp_t + h0_old @ W_hh + b_hh) ---
    {
      v8f acc = tile_gemm_lds8(h0_old + (size_t)rowg * RNN_H, slab_hh);
      const float* xpt = xp + ((size_t)t * RNN_B + rowg) * RNN_H;
#pragma unroll
      for (int v = 0; v < 8; ++v) {
        int m = v + 8 * half;
        float val = acc[v] + xpt[(size_t)m * RNN_H + colg] + bias_hh;
        h0_new[(size_t)(rowg + m) * RNN_H + colg] = tanhf(val);
      }
    }
    grid_barrier(bar, REC_BLOCKS);

    // --- layer 1: h1 = tanh(h0_new @ W_l + h1_old @ W_r + b_l + b_r) ---
    {
      v8f acc = tile_gemm2_lds8(h0_new + (size_t)rowg * RNN_H, slab_l,
                                h1_old + (size_t)rowg * RNN_H, slab_r);
#pragma unroll
      for (int v = 0; v < 8; ++v) {
        int m = v + 8 * half;
        h1_new[(size_t)(rowg + m) * RNN_H + colg] = tanhf(acc[v] + bias_lr);
      }
    }
    grid_barrier(bar, REC_BLOCKS);

    // --- output: y_t = h1_new @ W_ho + b_ho  -> out[b, t, :] ---
    {
      v8f acc = tile_gemm_lds8(h1_new + (size_t)rowg * RNN_H, slab_ho);
#pragma unroll
      for (int v = 0; v < 8; ++v) {
        int m = v + 8 * half;
        out[((size_t)(rowg + m) * RNN_T + t) * RNN_H + colg] = acc[v] + bias_ho;
      }
    }
    // Ping-pong buffers + the two barriers above already order every writer
    // after the last reader of its buffer; no third barrier needed.
  }
}

// ---------------------------------------------------------------------------
extern "C" void kernel_launch(void* const* d_in, const int* in_sizes, int n_in,
                              void* d_out, int out_size, void* d_ws, size_t ws_size,
                              hipStream_t stream) {
  const float* emb = (const float*)d_in[0];
  const float* Wih = (const float*)d_in[1];
  const float* bih = (const float*)d_in[2];
  const float* Whh = (const float*)d_in[3];
  const float* bhh = (const float*)d_in[4];
  const float* Wl  = (const float*)d_in[5];
  const float* bl  = (const float*)d_in[6];
  const float* Wr  = (const float*)d_in[7];
  const float* br  = (const float*)d_in[8];
  const float* Who = (const float*)d_in[9];
  const float* bho = (const float*)d_in[10];
  float* out = (float*)d_out;

  // Workspace layout:
  //   [0, T*B*H*4)        x_proj, [T,B,H] fp32 (128 MB)
  //   then 4*B*H fp32     h0 ping/pong + h1 ping/pong (1 MB)
  //   then 256B-aligned   barrier state (arrive @ +0, gen @ +256B)
  char* ws = (char*)d_ws;
  const size_t xp_bytes = (size_t)RNN_T * RNN_B * RNN_H * sizeof(float);
  const size_t h_bytes  = 4ull * RNN_B * RNN_H * sizeof(float);
  float* xp    = (float*)ws;
  float* hbufs = (float*)(ws + xp_bytes);
  float* h0buf = hbufs;
  float* h1buf = hbufs + 2ull * RNN_B * RNN_H;
  size_t bar_off = (xp_bytes + h_bytes + 255ull) & ~255ull;
  unsigned* bar = (unsigned*)(ws + bar_off);

  // 1) zero hidden state + barrier (re-run every launch: graph-safe)
  hipLaunchKernelGGL(rnn_init_kernel, dim3(1024), dim3(256), 0, stream,
                     hbufs, bar);

  // 2) input projection GEMM: 2048*64 tiles / 8 waves per block
  hipLaunchKernelGGL(rnn_xproj_kernel, dim3(16384), dim3(256), 0, stream,
                     emb, Wih, bih, xp);

  // 3) persistent recurrence: 64 blocks x 128 threads, 257 KB LDS each
  hipLaunchKernelGGL(rnn_recurrent_kernel, dim3(REC_BLOCKS), dim3(128),
                     REC_LDS_BYTES, stream,
                     xp, Whh, bhh, Wl, bl, Wr, br, Who, bho,
                     h0buf, h1buf, bar, out);
}